// PrimitiveModule_85959475462706
// MI455X (gfx1250) — compile-verified
//
#include <hip/hip_runtime.h>
#include <math.h>

// ---------------- problem constants ----------------
#define BB 16
#define KK 4096
#define CC 256
#define PP 1024
#define SS 16
#define HID 128
#define K0 259          // 3 + C... no: 3 + 256 = 259
#define KP0 260         // padded to multiple of 4
#define OUTCH 22        // 3 + 1 + 18
#define RADIUS_F 0.3f
#define EPS_F 1e-5f

typedef float v2f __attribute__((ext_vector_type(2)));
typedef float v8f __attribute__((ext_vector_type(8)));

// D = A(16x4) * B(4x16) + C, f32 WMMA.
__device__ __forceinline__ v8f wmma4(v2f a, v2f b, v8f c) {
    return __builtin_amdgcn_wmma_f32_16x16x4_f32(false, a, false, b, (short)0, c, false, false);
}

// ---------------- zero small stats region ----------------
__global__ void zero_kernel(float* p, int n) {
    int i = blockIdx.x * blockDim.x + threadIdx.x;
    if (i < n) p[i] = 0.f;
}

// ---------------- transpose features (B,C,K) -> (B,K,C) ----------------
__global__ void transpose_kernel(const float* __restrict__ f, float* __restrict__ ft) {
    __shared__ float tile[32][33];
    int b = blockIdx.z;
    int k0 = blockIdx.x * 32, c0 = blockIdx.y * 32;
    int tx = threadIdx.x, ty = threadIdx.y;     // 32 x 8
    for (int i = ty; i < 32; i += 8)
        tile[i][tx] = f[(size_t)b * CC * KK + (size_t)(c0 + i) * KK + (k0 + tx)];
    __syncthreads();
    for (int i = ty; i < 32; i += 8)
        ft[(size_t)b * KK * CC + (size_t)(k0 + i) * CC + (c0 + tx)] = tile[tx][i];
}

// ---------------- farthest point sampling: one block per batch ----------------
__global__ void fps_kernel(const float* __restrict__ xyz, int* __restrict__ fps_idx,
                           float* __restrict__ new_xyz) {
    __shared__ float px[KK], py[KK], pz[KK];
    __shared__ float redV[256];
    __shared__ int   redI[256];
    __shared__ int   curIdx;
    int b = blockIdx.x, tid = threadIdx.x;      // 256 threads
    for (int i = tid; i < KK; i += 256) {
        px[i] = xyz[(size_t)b * KK * 3 + i * 3 + 0];
        py[i] = xyz[(size_t)b * KK * 3 + i * 3 + 1];
        pz[i] = xyz[(size_t)b * KK * 3 + i * 3 + 2];
    }
    float mind[16];
#pragma unroll
    for (int j = 0; j < 16; ++j) mind[j] = 1e10f;
    if (tid == 0) {
        fps_idx[(size_t)b * PP] = 0;
        curIdx = 0;
    }
    __syncthreads();
    if (tid == 0) {
        new_xyz[((size_t)b * PP) * 3 + 0] = px[0];
        new_xyz[((size_t)b * PP) * 3 + 1] = py[0];
        new_xyz[((size_t)b * PP) * 3 + 2] = pz[0];
    }
    int last = 0;
    for (int it = 1; it < PP; ++it) {
        float lx = px[last], ly = py[last], lz = pz[last];
        float bv = -1.f; int bi = KK;
#pragma unroll
        for (int j = 0; j < 16; ++j) {
            int pt = tid * 16 + j;
            float dx = px[pt] - lx, dy = py[pt] - ly, dz = pz[pt] - lz;
            float d = dx * dx + dy * dy + dz * dz;
            float m = fminf(mind[j], d);
            mind[j] = m;
            if (m > bv) { bv = m; bi = pt; }    // first max within thread (pt increasing)
        }
        redV[tid] = bv; redI[tid] = bi;
        __syncthreads();
        for (int s = 128; s > 0; s >>= 1) {
            if (tid < s) {
                float v2 = redV[tid + s]; int i2 = redI[tid + s];
                if (v2 > redV[tid] || (v2 == redV[tid] && i2 < redI[tid])) {
                    redV[tid] = v2; redI[tid] = i2;
                }
            }
            __syncthreads();
        }
        if (tid == 0) {
            int nx = redI[0];
            curIdx = nx;
            fps_idx[(size_t)b * PP + it] = nx;
            new_xyz[((size_t)b * PP + it) * 3 + 0] = px[nx];
            new_xyz[((size_t)b * PP + it) * 3 + 1] = py[nx];
            new_xyz[((size_t)b * PP + it) * 3 + 2] = pz[nx];
        }
        __syncthreads();
        last = curIdx;
    }
}

// ---------------- ball query: one wave32 per query ----------------
__global__ void ballq_kernel(const float* __restrict__ xyz, const float* __restrict__ new_xyz,
                             int* __restrict__ gidx) {
    int q = blockIdx.x * 4 + (threadIdx.x >> 5);
    int lane = threadIdx.x & 31;
    int b = q >> 10;                            // q / PP
    float nx = new_xyz[(size_t)q * 3 + 0];
    float ny = new_xyz[(size_t)q * 3 + 1];
    float nz = new_xyz[(size_t)q * 3 + 2];
    const float r2 = RADIUS_F * RADIUS_F;
    int myIdx = KK - 1;                         // covers the "no neighbor" case
    int first = KK - 1;
    int cnt = 0;
    for (int base = 0; base < KK && cnt < SS; base += 32) {
        int pt = base + lane;
        const float* pr = xyz + ((size_t)b * KK + pt) * 3;
        float dx = pr[0] - nx, dy = pr[1] - ny, dz = pr[2] - nz;
        float d2 = dx * dx + dy * dy + dz * dz;
        unsigned mask = (unsigned)__ballot(d2 < r2);
        while (mask && cnt < SS) {
            int bit = __builtin_ctz(mask);
            mask &= mask - 1;
            int pidx = base + bit;
            if (cnt == 0) first = pidx;
            if (lane == cnt) myIdx = pidx;
            ++cnt;
        }
    }
    if (cnt > 0 && lane >= cnt && lane < SS) myIdx = first;
    if (lane < SS) gidx[(size_t)q * SS + lane] = myIdx;
}

// ---------------- layer0: gather [gxyz|gfeat] and GEMM (K=259) via WMMA ----------------
// 2 waves per block; wave = one query = one M=16 tile; 8 N-tiles of 16.
__global__ void layer0_kernel(const float* __restrict__ xyz, const float* __restrict__ featsT,
                              const float* __restrict__ new_xyz, const int* __restrict__ gidx,
                              const float* __restrict__ w0, float* __restrict__ X0,
                              float* __restrict__ sSum, float* __restrict__ sSq) {
    __shared__ float apan[2][16 * KP0];
    __shared__ float wt[16 * KP0];
    __shared__ float ls[HID], lq[HID];
    int tid = threadIdx.x, w = tid >> 5, lane = tid & 31;
    for (int i = tid; i < HID; i += 64) { ls[i] = 0.f; lq[i] = 0.f; }
    int q = blockIdx.x * 2 + w;
    int b = q >> 10;
    float nx0 = new_xyz[(size_t)q * 3 + 0];
    float nx1 = new_xyz[(size_t)q * 3 + 1];
    float nx2 = new_xyz[(size_t)q * 3 + 2];
    float* ap = apan[w];
    for (int s = 0; s < SS; ++s) {
        int idx = gidx[(size_t)q * SS + s];
        const float* fr = featsT + ((size_t)b * KK + idx) * CC;
        const float* xr = xyz + ((size_t)b * KK + idx) * 3;
        float* row = ap + s * KP0;
        for (int c = lane; c < KP0; c += 32) {
            float v;
            if (c < 3)        v = (xr[c] - (c == 0 ? nx0 : (c == 1 ? nx1 : nx2))) / RADIUS_F;
            else if (c < K0)  v = fr[c - 3];
            else              v = 0.f;
            row[c] = v;
        }
    }
    int half = lane >> 4, nl = lane & 15;
    int aoff = nl * KP0 + 2 * half;
    for (int t = 0; t < 8; ++t) {
        __syncthreads();
        for (int e = tid; e < 16 * KP0; e += 64) {
            int n = e / KP0, c = e - n * KP0;
            wt[e] = (c < K0) ? w0[(size_t)(t * 16 + n) * K0 + c] : 0.f;
        }
        __syncthreads();
        v8f acc = {};
        const float* aP = ap + aoff;
        const float* bP = wt + aoff;
        for (int kk = 0; kk < KP0; kk += 4) {
            v2f a  = *(const v2f*)(aP + kk);
            v2f bm = *(const v2f*)(bP + kk);
            acc = wmma4(a, bm, acc);
        }
        float s1 = 0.f, s2 = 0.f;
        size_t obase = (size_t)q * SS * HID + t * 16 + nl;
#pragma unroll
        for (int j = 0; j < 8; ++j) {
            float v = acc[j];
            X0[obase + (size_t)(j + 8 * half) * HID] = v;
            s1 += v; s2 += v * v;
        }
        s1 += __shfl_xor(s1, 16);
        s2 += __shfl_xor(s2, 16);
        if (lane < 16) { atomicAdd(&ls[t * 16 + nl], s1); atomicAdd(&lq[t * 16 + nl], s2); }
    }
    __syncthreads();
    for (int i = tid; i < HID; i += 64) { atomicAdd(&sSum[i], ls[i]); atomicAdd(&sSq[i], lq[i]); }
}

// ---------------- generic 128-in / 128-out GEMM with fused input BN+ReLU ----------------
// 4 waves per block, each wave = one row tile of 16 consecutive rows.
__global__ void gemm128_kernel(const float* __restrict__ Xin, const float* __restrict__ W,
                               const float* __restrict__ inS, const float* __restrict__ inB,
                               const float* __restrict__ outBias, float* __restrict__ Xout,
                               float* __restrict__ sSum, float* __restrict__ sSq) {
    __shared__ float apan[4][16 * HID];
    __shared__ float wt[16 * HID];
    __shared__ float ls[HID], lq[HID];
    int tid = threadIdx.x, w = tid >> 5, lane = tid & 31;
    ls[tid] = 0.f; lq[tid] = 0.f;
    size_t r0 = ((size_t)blockIdx.x * 4 + w) * 16;
    float* ap = apan[w];
    const float* xrow = Xin + r0 * HID;         // 16 consecutive rows are contiguous
    bool doAct = (inS != nullptr);
    for (int e = lane; e < 16 * HID; e += 32) {
        int c = e & (HID - 1);
        float v = xrow[e];
        if (doAct) v = fmaxf(v * inS[c] + inB[c], 0.f);
        ap[e] = v;
    }
    int half = lane >> 4, nl = lane & 15;
    int aoff = nl * HID + 2 * half;
    for (int t = 0; t < 8; ++t) {
        __syncthreads();
        for (int e = tid; e < 16 * HID; e += 128) wt[e] = W[(size_t)t * 16 * HID + e];
        __syncthreads();
        v8f acc = {};
        const float* aP = ap + aoff;
        const float* bP = wt + aoff;
        for (int kk = 0; kk < HID; kk += 4) {
            v2f a  = *(const v2f*)(aP + kk);
            v2f bm = *(const v2f*)(bP + kk);
            acc = wmma4(a, bm, acc);
        }
        float ob = outBias ? outBias[t * 16 + nl] : 0.f;
        float s1 = 0.f, s2 = 0.f;
        size_t obase = r0 * HID + t * 16 + nl;
#pragma unroll
        for (int j = 0; j < 8; ++j) {
            float v = acc[j] + ob;
            Xout[obase + (size_t)(j + 8 * half) * HID] = v;
            s1 += v; s2 += v * v;
        }
        s1 += __shfl_xor(s1, 16);
        s2 += __shfl_xor(s2, 16);
        if (lane < 16) { atomicAdd(&ls[t * 16 + nl], s1); atomicAdd(&lq[t * 16 + nl], s2); }
    }
    __syncthreads();
    atomicAdd(&sSum[tid], ls[tid]);
    atomicAdd(&sSq[tid], lq[tid]);
}

// ---------------- BN finalize: scale = g*rsqrt(var+eps), bias = be - mean*scale ----------------
__global__ void bnfin_kernel(const float* __restrict__ sum, const float* __restrict__ sq,
                             const float* __restrict__ g, const float* __restrict__ be,
                             float invN, float* __restrict__ scale, float* __restrict__ bias) {
    int i = threadIdx.x;
    if (i < HID) {
        float m = sum[i] * invN;
        float v = sq[i] * invN - m * m;
        float inv = rsqrtf(v + EPS_F);
        float sc = g[i] * inv;
        scale[i] = sc;
        bias[i]  = be[i] - m * sc;
    }
}

// ---------------- BN+ReLU + max over 16 samples ----------------
__global__ void maxpool_kernel(const float* __restrict__ X2, const float* __restrict__ sc,
                               const float* __restrict__ bi, float* __restrict__ agg) {
    int q = blockIdx.x, c = threadIdx.x;        // 128 threads
    const float* xp = X2 + (size_t)q * SS * HID + c;
    float s = sc[c], b = bi[c], m = 0.f;
    for (int i = 0; i < SS; ++i) {
        float v = fmaxf(xp[(size_t)i * HID] * s + b, 0.f);
        m = fmaxf(m, v);
    }
    agg[(size_t)q * HID + c] = m;
}

// ---------------- final head: out = relu(bn(t2)) @ w3^T + b3, center offset fused ----------------
__global__ void final_kernel(const float* __restrict__ t2, const float* __restrict__ sc,
                             const float* __restrict__ bi, const float* __restrict__ w3,
                             const float* __restrict__ b3, const float* __restrict__ new_xyz,
                             float* __restrict__ out) {
    __shared__ float row[HID];
    int q = blockIdx.x, l = threadIdx.x;        // 32 threads
    for (int c = l; c < HID; c += 32)
        row[c] = fmaxf(t2[(size_t)q * HID + c] * sc[c] + bi[c], 0.f);
    __syncthreads();
    if (l < OUTCH) {
        const float* wr = w3 + (size_t)l * HID;
        float acc = b3[l];
        for (int c = 0; c < HID; ++c) acc += row[c] * wr[c];
        if (l < 3) acc += new_xyz[(size_t)q * 3 + l];
        out[(size_t)q * OUTCH + l] = acc;
    }
}

// ---------------- host ----------------
extern "C" void kernel_launch(void* const* d_in, const int* in_sizes, int n_in,
                              void* d_out, int out_size, void* d_ws, size_t ws_size,
                              hipStream_t stream) {
    const float* xyz      = (const float*)d_in[0];
    const float* features = (const float*)d_in[1];
    const float* mlp_w0   = (const float*)d_in[2];
    const float* mlp_g0   = (const float*)d_in[3];
    const float* mlp_be0  = (const float*)d_in[4];
    const float* mlp_w1   = (const float*)d_in[5];
    const float* mlp_g1   = (const float*)d_in[6];
    const float* mlp_be1  = (const float*)d_in[7];
    const float* mlp_w2   = (const float*)d_in[8];
    const float* mlp_g2   = (const float*)d_in[9];
    const float* mlp_be2  = (const float*)d_in[10];
    const float* w1       = (const float*)d_in[11];
    const float* b1       = (const float*)d_in[12];
    const float* g1       = (const float*)d_in[13];
    const float* be1      = (const float*)d_in[14];
    const float* w2       = (const float*)d_in[15];
    const float* b2       = (const float*)d_in[16];
    const float* g2       = (const float*)d_in[17];
    const float* be2      = (const float*)d_in[18];
    const float* w3       = (const float*)d_in[19];
    const float* b3       = (const float*)d_in[20];
    float* out = (float*)d_out;

    char* ws = (char*)d_ws;
    size_t off = 0;
    auto alloc = [&](size_t bytes) -> char* {
        char* p = ws + off;
        off += (bytes + 255) & ~(size_t)255;
        return p;
    };
    const size_t NQ = (size_t)BB * PP;              // 16384 queries
    const size_t NR = NQ * SS;                      // 262144 grouped rows
    float* featsT  = (float*)alloc(sizeof(float) * BB * KK * CC);
    float* X0      = (float*)alloc(sizeof(float) * NR * HID);
    float* X1      = (float*)alloc(sizeof(float) * NR * HID);
    float* agg     = (float*)alloc(sizeof(float) * NQ * HID);
    float* t1      = (float*)alloc(sizeof(float) * NQ * HID);
    float* t2      = (float*)alloc(sizeof(float) * NQ * HID);
    float* new_xyz = (float*)alloc(sizeof(float) * NQ * 3);
    int*   fps_idx = (int*)alloc(sizeof(int) * NQ);
    int*   gidx    = (int*)alloc(sizeof(int) * NQ * SS);
    float* stats   = (float*)alloc(sizeof(float) * 5 * 512);
    // per-layer stats layout: +0 sum[128], +128 sumsq[128], +256 scale[128], +384 bias[128]
    float* st[5];
    for (int l = 0; l < 5; ++l) st[l] = stats + l * 512;

    zero_kernel<<<(5 * 256 + 255) / 256, 256, 0, stream>>>(stats, 5 * 256);
    transpose_kernel<<<dim3(KK / 32, CC / 32, BB), dim3(32, 8), 0, stream>>>(features, featsT);
    fps_kernel<<<BB, 256, 0, stream>>>(xyz, fps_idx, new_xyz);
    ballq_kernel<<<NQ / 4, 128, 0, stream>>>(xyz, new_xyz, gidx);

    // grouped MLP layer 0 (gather + 259->128)
    layer0_kernel<<<NQ / 2, 64, 0, stream>>>(xyz, featsT, new_xyz, gidx, mlp_w0,
                                             X0, st[0] + 0, st[0] + 128);
    bnfin_kernel<<<1, 128, 0, stream>>>(st[0], st[0] + 128, mlp_g0, mlp_be0,
                                        1.0f / (float)NR, st[0] + 256, st[0] + 384);
    // grouped layer 1
    gemm128_kernel<<<NR / 16 / 4, 128, 0, stream>>>(X0, mlp_w1, st[0] + 256, st[0] + 384,
                                                    nullptr, X1, st[1] + 0, st[1] + 128);
    bnfin_kernel<<<1, 128, 0, stream>>>(st[1], st[1] + 128, mlp_g1, mlp_be1,
                                        1.0f / (float)NR, st[1] + 256, st[1] + 384);
    // grouped layer 2 (reuse X0 as output)
    gemm128_kernel<<<NR / 16 / 4, 128, 0, stream>>>(X1, mlp_w2, st[1] + 256, st[1] + 384,
                                                    nullptr, X0, st[2] + 0, st[2] + 128);
    bnfin_kernel<<<1, 128, 0, stream>>>(st[2], st[2] + 128, mlp_g2, mlp_be2,
                                        1.0f / (float)NR, st[2] + 256, st[2] + 384);
    // BN+ReLU + max over samples
    maxpool_kernel<<<NQ, 128, 0, stream>>>(X0, st[2] + 256, st[2] + 384, agg);
    // FC1 (input already activated)
    gemm128_kernel<<<NQ / 16 / 4, 128, 0, stream>>>(agg, w1, nullptr, nullptr,
                                                    b1, t1, st[3] + 0, st[3] + 128);
    bnfin_kernel<<<1, 128, 0, stream>>>(st[3], st[3] + 128, g1, be1,
                                        1.0f / (float)NQ, st[3] + 256, st[3] + 384);
    // FC2
    gemm128_kernel<<<NQ / 16 / 4, 128, 0, stream>>>(t1, w2, st[3] + 256, st[3] + 384,
                                                    b2, t2, st[4] + 0, st[4] + 128);
    bnfin_kernel<<<1, 128, 0, stream>>>(st[4], st[4] + 128, g2, be2,
                                        1.0f / (float)NQ, st[4] + 256, st[4] + 384);
    // head + center fuse
    final_kernel<<<NQ, 32, 0, stream>>>(t2, st[4] + 256, st[4] + 384, w3, b3, new_xyz, out);

    (void)in_sizes; (void)n_in; (void)out_size; (void)ws_size;
}